// GCN_7215545057463
// MI455X (gfx1250) — compile-verified
//
#include <hip/hip_runtime.h>
#include <hip/hip_bf16.h>
#include <math.h>

typedef __attribute__((ext_vector_type(2))) float v2f;
typedef __attribute__((ext_vector_type(8))) float v8f;

#define CCH   64      // channels (C_IN == C_OUT == 64)
#define FILLW 2.0f    // improved-GCN self-loop weight

// Native f32 atomic add (global_atomic_add_f32), relaxed/agent scope.
__device__ __forceinline__ void atomic_add_f32(float* p, float v) {
  __hip_atomic_fetch_add(p, v, __ATOMIC_RELAXED, __HIP_MEMORY_SCOPE_AGENT);
}

// ---------------------------------------------------------------------------
// 1) zero the output accumulator, seed deg[n] with the self-loop fill weight
// ---------------------------------------------------------------------------
__global__ void gcn_init(float* __restrict__ out, float* __restrict__ deg,
                         int total, int n_nodes) {
  int t = blockIdx.x * blockDim.x + threadIdx.x;
  if (t < total)   out[t] = 0.0f;
  if (t < n_nodes) deg[t] = FILLW;
}

// ---------------------------------------------------------------------------
// 2) deg[dst] += w  over all edges
// ---------------------------------------------------------------------------
__global__ void gcn_deg(const int* __restrict__ dst, const float* __restrict__ w,
                        float* __restrict__ deg, int n_edges) {
  int e = blockIdx.x * blockDim.x + threadIdx.x;
  if (e < n_edges) atomic_add_f32(&deg[dst[e]], w[e]);
}

// ---------------------------------------------------------------------------
// 3) deg -> dinv = rsqrt(deg) (in place)
// ---------------------------------------------------------------------------
__global__ void gcn_dinv(float* __restrict__ deg, int n_nodes) {
  int n = blockIdx.x * blockDim.x + threadIdx.x;
  if (n < n_nodes) {
    float d = deg[n];
    deg[n] = (d > 0.0f) ? rsqrtf(d) : 0.0f;
  }
}

// ---------------------------------------------------------------------------
// 4) XW = X @ W via V_WMMA_F32_16X16X4_F32, one 16x16 tile per wave32.
//    A(16x4 f32): lanes 0-15 hold K={k0,k0+1}, lanes 16-31 hold K={k0+2,k0+3}
//    B(4x16 f32): mirrored across columns
//    C/D(16x16):  VGPR i -> row i + 8*(lane>=16), col = lane%16
// ---------------------------------------------------------------------------
__global__ void gcn_gemm_wmma(const float* __restrict__ X,
                              const float* __restrict__ W,
                              float* __restrict__ XW, int n_nodes) {
  const int wave = (blockIdx.x * blockDim.x + threadIdx.x) >> 5;
  const int lane = threadIdx.x & 31;
  const int n_row_tiles = (n_nodes + 15) >> 4;
  const int row_tile = wave >> 2;          // 4 column tiles of 16 (CCH = 64)
  const int col_tile = wave & 3;
  if (row_tile >= n_row_tiles) return;     // wave-uniform: EXEC stays all-ones

  const int half = lane >> 4;              // 0: lanes 0-15, 1: lanes 16-31
  const int l16  = lane & 15;
  int row = row_tile * 16 + l16;
  int ldrow = (row < n_nodes) ? row : (n_nodes - 1);   // clamp loads (N%16==0 anyway)

  const float* __restrict__ xrow = X + (size_t)ldrow * CCH;
  const float* __restrict__ wcol = W + col_tile * 16 + l16;

  v8f acc = {};
#pragma unroll
  for (int k0 = 0; k0 < CCH; k0 += 4) {
    const int ka = k0 + 2 * half;
    v2f a, b;
    a.x = xrow[ka + 0];
    a.y = xrow[ka + 1];
    b.x = wcol[(size_t)(ka + 0) * CCH];
    b.y = wcol[(size_t)(ka + 1) * CCH];
    acc = __builtin_amdgcn_wmma_f32_16x16x4_f32(
        /*neg_a=*/false, a, /*neg_b=*/false, b,
        /*c_mod=*/(short)0, acc, /*reuse_a=*/false, /*reuse_b=*/false);
  }

  const int base_row = row_tile * 16 + 8 * half;
  float* __restrict__ dstp = XW + (size_t)base_row * CCH + col_tile * 16 + l16;
#pragma unroll
  for (int i = 0; i < 8; ++i) {
    if (base_row + i < n_nodes) dstp[(size_t)i * CCH] = acc[i];
  }
}

// ---------------------------------------------------------------------------
// 5) one wave32 per edge: out[dst] += xw[src] * (dinv[src]*w*dinv[dst])
//    lane handles channels {lane, lane+32} -> two coalesced 128B rows/edge
// ---------------------------------------------------------------------------
__global__ void gcn_scatter(const int* __restrict__ src, const int* __restrict__ dst,
                            const float* __restrict__ w, const float* __restrict__ dinv,
                            const float* __restrict__ xw, float* __restrict__ out,
                            int n_edges) {
  const int gt   = blockIdx.x * blockDim.x + threadIdx.x;
  const int e    = gt >> 5;
  const int lane = gt & 31;
  if (e >= n_edges) return;

  const int s = src[e];
  const int d = dst[e];
  float norm = 0.0f;
  if (lane == 0) norm = dinv[s] * w[e] * dinv[d];
  norm = __shfl(norm, 0, 32);

  const float* __restrict__ xr = xw + (size_t)s * CCH;
  float* __restrict__ orow     = out + (size_t)d * CCH;

  atomic_add_f32(&orow[lane],      xr[lane]      * norm);
  atomic_add_f32(&orow[lane + 32], xr[lane + 32] * norm);
}

// ---------------------------------------------------------------------------
// 6) epilogue: out = tanh(out + xw * (FILL * dinv^2) + b)   (self-loop term)
// ---------------------------------------------------------------------------
__global__ void gcn_final(float* __restrict__ out, const float* __restrict__ xw,
                          const float* __restrict__ dinv, const float* __restrict__ b,
                          int total) {
  int t = blockIdx.x * blockDim.x + threadIdx.x;
  if (t < total) {
    int n = t >> 6;          // / CCH
    int c = t & (CCH - 1);
    float di = dinv[n];
    float v  = out[t] + xw[t] * (FILLW * di * di) + b[c];
    out[t] = tanhf(v);
  }
}

// ---------------------------------------------------------------------------
extern "C" void kernel_launch(void* const* d_in, const int* in_sizes, int n_in,
                              void* d_out, int out_size, void* d_ws, size_t ws_size,
                              hipStream_t stream) {
  const float* x  = (const float*)d_in[0];   // [N, 64]
  const int*   ei = (const int*)d_in[1];     // [2, E] row-major
  const float* ew = (const float*)d_in[2];   // [E]
  const float* W  = (const float*)d_in[3];   // [64, 64]
  const float* b  = (const float*)d_in[4];   // [64]
  float* out = (float*)d_out;

  const int n_nodes = in_sizes[0] / CCH;
  const int n_edges = in_sizes[2];
  const int* src = ei;
  const int* dst = ei + n_edges;

  // workspace: xw [N*64 f32], then dinv [N f32]  (~26 MB)
  float* xw   = (float*)d_ws;
  float* dinv = xw + (size_t)n_nodes * CCH;

  const int total = n_nodes * CCH;
  const int B = 256;

  gcn_init<<<(total + B - 1) / B, B, 0, stream>>>(out, dinv, total, n_nodes);
  gcn_deg<<<(n_edges + B - 1) / B, B, 0, stream>>>(dst, ew, dinv, n_edges);
  gcn_dinv<<<(n_nodes + B - 1) / B, B, 0, stream>>>(dinv, n_nodes);

  const int n_row_tiles = (n_nodes + 15) / 16;
  const int waves = n_row_tiles * (CCH / 16);          // 25000 waves
  gcn_gemm_wmma<<<(waves + 7) / 8, B, 0, stream>>>(x, W, xw, n_nodes);

  const long long sthreads = (long long)n_edges * 32;  // 51.2M threads
  gcn_scatter<<<(unsigned)((sthreads + B - 1) / B), B, 0, stream>>>(
      src, dst, ew, dinv, xw, out, n_edges);

  gcn_final<<<(total + B - 1) / B, B, 0, stream>>>(out, xw, dinv, b, total);
}